// SpectralShiftTracker_56126632624478
// MI455X (gfx1250) — compile-verified
//
#include <hip/hip_runtime.h>
#include <math.h>

// SpectralShiftTracker: eta[b] = tanh( logdet( X_b^T X_b / N + eps*I ) / D )
// X: [B=256, N=1024, D=256] fp32.  One workgroup per batch:
//   phase 1: SYRK via V_WMMA_F32_16X16X4_F32, X staged transposed in LDS,
//            double-buffered + register-pipelined global->LDS staging
//   phase 2: in-LDS pivot-free LU (SPD) -> logdet = sum log(U_jj)
//   phase 3: tanh(logdet/D) -> out[b]

#define DIMD   256
#define NSAMP  1024
#define CHUNK  16                 // K rows staged per pipeline stage
#define NCHUNK (NSAMP / CHUNK)    // 64
#define KSTR   18                 // padded k-stride (dwords): even (b64 align), conflict-free
#define CSTR   257                // padded C row stride (dwords)
#define EPS    1e-6f

typedef float v2f __attribute__((ext_vector_type(2)));
typedef float v8f __attribute__((ext_vector_type(8)));

extern "C" __global__ __launch_bounds__(1024, 1)
void spectral_shift_kernel(const float* __restrict__ X, float* __restrict__ out) {
    extern __shared__ float smem[];
    float* xs = smem;                           // 2 * DIMD * KSTR floats (double-buffered chunk)
    float* Cs = smem + 2 * DIMD * KSTR;         // DIMD * CSTR floats (covariance / LU workspace)

    const int tid  = threadIdx.x;               // 0..1023 (32 waves, wave32)
    const int b    = blockIdx.x;
    const int wave = tid >> 5;
    const int lane = tid & 31;
    const int lm   = lane & 15;                 // M/N position within 16x16 tile
    const int khalf = (lane >> 4) << 1;         // lanes 16-31 hold K+2,K+3

    // tile ownership: wave owns 8 tiles, t = wave*8 + r ; ti = wave>>1, tj = (wave&1)*8 + r
    const int ti  = wave >> 1;
    const int tj0 = (wave & 1) * 8;

    // per-lane dword offsets into a chunk buffer for A frag and the 8 B frags
    const int aoff = (ti * 16 + lm) * KSTR + khalf;
    int boff[8];
#pragma unroll
    for (int r = 0; r < 8; ++r)
        boff[r] = ((tj0 + r) * 16 + lm) * KSTR + khalf;

    v8f acc[8];
#pragma unroll
    for (int r = 0; r < 8; ++r) acc[r] = (v8f)(0.0f);

    const float* __restrict__ Xb = X + (size_t)b * NSAMP * DIMD;

    // staging thread mapping: d = tid%256, base row k0 = tid/256 (0..3), 4 elems/thread
    const int d    = tid & 255;
    const int k0   = tid >> 8;
    const int sdst = d * KSTR + k0;             // within-buffer LDS float index

    // ---- prologue: stage chunk 0 into buffer 0 ----
    float regs[4];
#pragma unroll
    for (int it = 0; it < 4; ++it)
        regs[it] = Xb[(size_t)(k0 + it * 4) * DIMD + d];
#pragma unroll
    for (int it = 0; it < 4; ++it)
        xs[sdst + it * 4] = regs[it];
    __syncthreads();

    // ---- pipelined main loop: one barrier per chunk ----
    for (int c = 0; c < NCHUNK; ++c) {
        const int pbuf = (c & 1) * (DIMD * KSTR);

        // (1) issue next chunk's global loads early; latency overlaps WMMA block
        if (c + 1 < NCHUNK) {
            const float* src = Xb + (size_t)((c + 1) * CHUNK + k0) * DIMD + d;
#pragma unroll
            for (int it = 0; it < 4; ++it)
                regs[it] = src[(size_t)it * 4 * DIMD];
            if (c + 2 < NCHUNK)
                __builtin_prefetch(Xb + (size_t)((c + 2) * CHUNK + k0) * DIMD + d, 0, 0);
        }

        // (2) SYRK on current buffer: 4 k-steps x 8 tiles
#pragma unroll
        for (int ks = 0; ks < CHUNK / 4; ++ks) {
            v2f a = *(const v2f*)&xs[pbuf + aoff + ks * 4];   // {K,K+1}/{K+2,K+3}
#pragma unroll
            for (int r = 0; r < 8; ++r) {
                v2f bf = *(const v2f*)&xs[pbuf + boff[r] + ks * 4];
                acc[r] = __builtin_amdgcn_wmma_f32_16x16x4_f32(
                    false, a, false, bf, (short)0, acc[r], false, false);
            }
        }

        // (3) commit next chunk into the other buffer (waits on its loads only here)
        if (c + 1 < NCHUNK) {
            const int nbuf = ((c + 1) & 1) * (DIMD * KSTR);
#pragma unroll
            for (int it = 0; it < 4; ++it)
                xs[nbuf + sdst + it * 4] = regs[it];
        }

        // (4) single barrier: orders buffer reuse across iterations
        __syncthreads();
    }

    // scatter C tiles to LDS with scale + eps on diagonal
    // C/D layout: VGPR p, lanes 0-15 -> (M=p, N=lane); lanes 16-31 -> (M=p+8, N=lane-16)
    const float scale = 1.0f / (float)NSAMP;
    const int mh = (lane >> 4) * 8;
#pragma unroll
    for (int r = 0; r < 8; ++r) {
        const int gj = (tj0 + r) * 16 + lm;
#pragma unroll
        for (int p = 0; p < 8; ++p) {
            const int gi = ti * 16 + mh + p;
            float v = acc[r][p] * scale;
            if (gi == gj) v += EPS;
            Cs[gi * CSTR + gj] = v;
        }
    }
    __syncthreads();

    // pivot-free right-looking LU on SPD matrix; diag becomes U_jj
    // wave handles rows i = j+1+wave (+32), lane handles cols k = j+1+lane (+32)
    for (int j = 0; j < DIMD - 1; ++j) {
        const float inv = 1.0f / Cs[j * CSTR + j];
        for (int i = j + 1 + wave; i < DIMD; i += 32) {
            const float lij = Cs[i * CSTR + j] * inv;
            for (int k = j + 1 + lane; k < DIMD; k += 32)
                Cs[i * CSTR + k] -= lij * Cs[j * CSTR + k];
        }
        __syncthreads();
    }

    // logdet = sum log(diag); then tanh(logdet / D)
    if (tid < DIMD) xs[tid] = logf(Cs[tid * CSTR + tid]);
    __syncthreads();
    if (tid == 0) {
        float s = 0.0f;
        for (int i = 0; i < DIMD; ++i) s += xs[i];
        out[b] = tanhf(s / (float)DIMD);
    }
}

extern "C" void kernel_launch(void* const* d_in, const int* in_sizes, int n_in,
                              void* d_out, int out_size, void* d_ws, size_t ws_size,
                              hipStream_t stream) {
    (void)in_sizes; (void)n_in; (void)d_ws; (void)ws_size; (void)out_size;
    const float* X = (const float*)d_in[0];
    float* out = (float*)d_out;

    const size_t shmem = (size_t)(2 * DIMD * KSTR + DIMD * CSTR) * sizeof(float); // ~300 KB
    static bool attr_set = false;
    if (!attr_set) {
        hipFuncSetAttribute((const void*)spectral_shift_kernel,
                            hipFuncAttributeMaxDynamicSharedMemorySize, (int)shmem);
        attr_set = true;
    }
    spectral_shift_kernel<<<256, 1024, shmem, stream>>>(X, out);
}